// MtMCreation_67353677135973
// MI455X (gfx1250) — compile-verified
//
#include <hip/hip_runtime.h>

typedef float v2f __attribute__((ext_vector_type(2)));
typedef float v8f __attribute__((ext_vector_type(8)));

namespace {
constexpr int Bc = 8, Hc = 96, Wc = 96, Oc = 8;
constexpr int NEQ   = 2 * Hc * Wc;   // 18432 equations per (b,o)
constexpr int WAVES = 8;             // 256-thread workgroup = 8 wave32
constexpr int LDSS  = 34;            // padded LDS row stride in floats (even -> b64 aligned,
                                     // 34*m mod 64 distinct for m=0..15 -> conflict-free)
}

// Kernel 1: per (b,o,slice) workgroup, accumulate a 16x16 f32 Gram tile per wave
// via v_wmma_f32_16x16x4_f32 rank-4 symmetric updates, then reduce the 8 wave
// tiles in f64 into one per-workgroup partial in d_ws.
__global__ __launch_bounds__(256) void mtm_partials(
    const float* __restrict__ obj, const float* __restrict__ wpix,
    const float* __restrict__ cm, const float* __restrict__ ck,
    double* __restrict__ ws, int split)
{
  const int bo    = blockIdx.x / split;
  const int slice = blockIdx.x - bo * split;
  const int b     = bo >> 3;    // Oc == 8
  const int o     = bo & 7;
  const int tid   = threadIdx.x;
  const int wave  = tid >> 5;
  const int lane  = tid & 31;
  const int half  = lane >> 4;  // 0: lanes 0-15, 1: lanes 16-31
  const int lm    = lane & 15;

  __shared__ float smem[WAVES * 16 * LDSS];  // 17408 B, reused for the f64 reduce
  for (int i = tid; i < WAVES * 16 * LDSS; i += 256) smem[i] = 0.0f;
  __syncthreads();
  float* wsm = smem + wave * 16 * LDSS;

  // Uniform per-(b) constants (TCR = identity folds A into a sparse pattern).
  const float fu = -cm[b * 9 + 0];
  const float fv = -cm[b * 9 + 4];
  const float cu =  cm[b * 9 + 2];
  const float cv =  cm[b * 9 + 5];
  const float uA = ck[b * 4 + 0], vA = ck[b * 4 + 1];
  const float uB = ck[b * 4 + 2], vB = ck[b * 4 + 3];

  const int eqPerWG   = NEQ / split;
  const int eqPerWave = eqPerWG / WAVES;
  const int iters     = eqPerWave >> 5;   // 32 equations per wave-iteration
  const int eqBase    = slice * eqPerWG + wave * eqPerWave;

  v8f acc = {};
  for (int it = 0; it < iters; ++it) {
    // ---- each lane builds one equation's 13-vector ----
    const int e   = eqBase + (it << 5) + lane;
    const int pix = e >> 1;
    const int r   = e & 1;
    const int h   = pix / Wc;
    const int w   = pix - h * Wc;
    const int base = (((b * Hc + h) * Wc + w) * Oc + o);
    const float p1 = obj[base * 3 + 0];
    const float p2 = obj[base * 3 + 1];
    const float p3 = obj[base * 3 + 2];
    const float wa = wpix[base * 4 + (r << 1) + 0];
    const float wb = wpix[base * 4 + (r << 1) + 1];
    const float du = (float)w * uA + uB - cu;
    const float dv = (float)h * vA + vB - cv;
    const float c1 = wa * fu;
    const float c2 = wb * fv;
    const float c3 = wa * du + wb * dv;
    // transposed store: smem[m][eq] so the WMMA read is a contiguous b64
    wsm[ 0 * LDSS + lane] = c1 * p1;
    wsm[ 1 * LDSS + lane] = c1 * p2;
    wsm[ 2 * LDSS + lane] = c1 * p3;
    wsm[ 3 * LDSS + lane] = c2 * p1;
    wsm[ 4 * LDSS + lane] = c2 * p2;
    wsm[ 5 * LDSS + lane] = c2 * p3;
    wsm[ 6 * LDSS + lane] = c3 * p1;
    wsm[ 7 * LDSS + lane] = c3 * p2;
    wsm[ 8 * LDSS + lane] = c3 * p3;
    wsm[10 * LDSS + lane] = c1;
    wsm[11 * LDSS + lane] = c2;
    wsm[12 * LDSS + lane] = c3;
    // rows 9, 13..15 stay zero (zeroed once above)

    // wave-local visibility: all 32 lanes' ds_stores land before transposed reads
    asm volatile("s_wait_dscnt 0" ::: "memory");

    // ---- 8 rank-4 symmetric updates; A and B register images are identical ----
#pragma unroll
    for (int kk = 0; kk < 8; ++kk) {
      const int e0 = (kk << 2) + (half << 1);   // even -> 8B aligned
      v2f ab = *reinterpret_cast<const v2f*>(&wsm[lm * LDSS + e0]);
      acc = __builtin_amdgcn_wmma_f32_16x16x4_f32(
          false, ab, false, ab, (short)0, acc, false, false);
    }
  }

  // ---- deterministic f64 reduction of the 8 wave tiles ----
  __syncthreads();
#pragma unroll
  for (int v = 0; v < 8; ++v) {
    const int M = v + (half << 3);              // C/D layout: lanes16-31 are rows M+8
    smem[wave * 256 + M * 16 + lm] = acc[v];
  }
  __syncthreads();
  double s = 0.0;
#pragma unroll
  for (int wv = 0; wv < WAVES; ++wv) s += (double)smem[wv * 256 + tid];
  ws[(size_t)blockIdx.x * 256 + tid] = s;
}

// Kernel 2: sum the per-slice f64 partials in fixed order, emit 13x13 per (b,o).
__global__ __launch_bounds__(256) void mtm_reduce(
    const double* __restrict__ ws, double* __restrict__ out, int split)
{
  const int bo = blockIdx.x;
  const int t  = threadIdx.x;
  double s = 0.0;
  for (int sl = 0; sl < split; ++sl)
    s += ws[(size_t)(bo * split + sl) * 256 + t];
  const int M = t >> 4, N = t & 15;
  if (M < 13 && N < 13)
    out[(size_t)bo * 169 + M * 13 + N] = s;
}

extern "C" void kernel_launch(void* const* d_in, const int* in_sizes, int n_in,
                              void* d_out, int out_size, void* d_ws, size_t ws_size,
                              hipStream_t stream) {
  const float* obj = (const float*)d_in[0];
  const float* wpx = (const float*)d_in[1];
  const float* cm  = (const float*)d_in[2];
  const float* ck  = (const float*)d_in[3];
  double* out = (double*)d_out;   // reference output dtype is float64
  double* ws  = (double*)d_ws;

  int split = 8;                  // candidates {8,4,2,1} all divide 72 wave-iterations
  const size_t perSplitBytes = (size_t)Bc * Oc * 256 * sizeof(double); // 128 KiB
  while (split > 1 && (size_t)split * perSplitBytes > ws_size) split >>= 1;

  mtm_partials<<<dim3(Bc * Oc * split), dim3(256), 0, stream>>>(obj, wpx, cm, ck, ws, split);
  mtm_reduce  <<<dim3(Bc * Oc),         dim3(256), 0, stream>>>(ws, out, split);
}